// AnalyticGaussianVPSScore_86887188398125
// MI455X (gfx1250) — compile-verified
//
#include <hip/hip_runtime.h>

#define NITER 64
#define BB 128   // batch
#define DD 512   // dim

typedef float v2f __attribute__((ext_vector_type(2)));
typedef float v8f __attribute__((ext_vector_type(8)));

// Per-batch alpha^2, sigma^2 and Chebyshev coefficient tables.
__global__ void coef_kernel(const float* __restrict__ t,
                            float* __restrict__ a2, float* __restrict__ s2,
                            float* __restrict__ c1t, float* __restrict__ c2t) {
    int i = threadIdx.x;
    if (i >= BB) return;
    float tv = t[i];
    float ib = 0.1f * tv + 0.5f * (20.0f - 0.1f) * tv * tv;
    float a  = expf(-0.5f * ib);
    float a2v = a * a;
    float s2v = fmaxf(1.0f - a2v, 1e-12f);
    a2[i] = a2v;
    s2[i] = s2v;
    // spectrum bounds of Sigma_i = a^2*cov0 + s^2*I, lambda(cov0) in [0.1, 4.6]
    float lmin = 0.1f * a2v + s2v;
    float lmax = 4.6f * a2v + s2v;
    float theta = 0.5f * (lmax + lmin);
    float delta = 0.5f * (lmax - lmin);
    float sig = theta / delta;
    float rho_prev = 1.0f / sig;
    for (int k = 1; k < NITER; ++k) {
        float rho = 1.0f / (2.0f * sig - rho_prev);
        c1t[k * BB + i] = rho * rho_prev;
        c2t[k * BB + i] = 2.0f * rho / delta;
        rho_prev = rho;
    }
}

// r0 = b = x_i - a*mean0 ; d0 = r0/theta ; x1 = d0.   Layout: [D][B], elem = d*BB + i.
__global__ void init_kernel(const float* __restrict__ x, const float* __restrict__ mean0,
                            const float* __restrict__ a2, const float* __restrict__ s2,
                            float* __restrict__ r, float* __restrict__ d0,
                            float* __restrict__ xs) {
    int idx = blockIdx.x * blockDim.x + threadIdx.x;   // 0 .. D*B-1
    int i = idx & (BB - 1);
    int d = idx >> 7;
    float a2v = a2[i], s2v = s2[i];
    float a = sqrtf(a2v);
    float b = x[i * DD + d] - a * mean0[d];
    float theta = 2.35f * a2v + s2v;                   // (4.6+0.1)/2 * a^2 + s^2
    r[idx] = b;
    float dv = b / theta;
    d0[idx] = dv;
    xs[idx] = dv;
}

// One Chebyshev step: T = cov0 @ dIn (f32 WMMA GEMM), then fused elementwise update:
//   Ad = a^2*T + s^2*dIn ; r -= Ad ; dOut = c1*dIn + c2*r ; xs += dOut
__global__ __launch_bounds__(128)
void cheb_iter_kernel(const float* __restrict__ cov0,
                      const float* __restrict__ dIn, float* __restrict__ dOut,
                      float* __restrict__ r, float* __restrict__ xs,
                      const float* __restrict__ a2, const float* __restrict__ s2,
                      const float* __restrict__ c1, const float* __restrict__ c2) {
    const int lane  = threadIdx.x & 31;
    const int wave  = threadIdx.x >> 5;
    const int l15   = lane & 15;
    const int lhalf = lane >> 4;
    const int m0 = blockIdx.x * 16;                    // 32 M-tiles
    const int n0 = (blockIdx.y * 4 + wave) * 16;       // 8 N-tiles

    v8f acc = {0.f, 0.f, 0.f, 0.f, 0.f, 0.f, 0.f, 0.f};
    // A fragment (16x4 f32): lanes 0-15 -> K = k,k+1 ; lanes 16-31 -> K = k+2,k+3
    const float* arow = cov0 + (m0 + l15) * DD + 2 * lhalf;
    // B fragment (4x16 f32): same K split, N across lanes 0-15
    const float* bcol = dIn + (2 * lhalf) * BB + n0 + l15;

#pragma unroll 4
    for (int k = 0; k < DD; k += 4) {
        v2f afrag = *(const v2f*)(arow + k);           // b64 load
        v2f bfrag;
        bfrag.x = bcol[k * BB];
        bfrag.y = bcol[k * BB + BB];
        acc = __builtin_amdgcn_wmma_f32_16x16x4_f32(false, afrag, false, bfrag,
                                                    (short)0, acc, false, false);
    }

    // C/D layout: VGPR j, lanes 0-15 -> M = j ; lanes 16-31 -> M = j+8 ; N = lane&15
    const int col = n0 + l15;
    const float a2c = a2[col], s2c = s2[col];
    const float c1c = c1[col], c2c = c2[col];
#pragma unroll
    for (int j = 0; j < 8; ++j) {
        int row = m0 + j + 8 * lhalf;
        int idx = row * BB + col;
        float din = dIn[idx];
        float Ad  = a2c * acc[j] + s2c * din;
        float rn  = r[idx] - Ad;
        r[idx] = rn;
        float dn = c1c * din + c2c * rn;
        dOut[idx] = dn;
        xs[idx] += dn;
    }
}

// out[i][d] = -xs[d][i]
__global__ void finalize_kernel(const float* __restrict__ xs, float* __restrict__ out) {
    int idx = blockIdx.x * blockDim.x + threadIdx.x;
    int i = idx & (BB - 1);
    int d = idx >> 7;
    out[i * DD + d] = -xs[idx];
}

extern "C" void kernel_launch(void* const* d_in, const int* in_sizes, int n_in,
                              void* d_out, int out_size, void* d_ws, size_t ws_size,
                              hipStream_t stream) {
    (void)in_sizes; (void)n_in; (void)out_size; (void)ws_size;
    const float* t     = (const float*)d_in[0];
    const float* x     = (const float*)d_in[1];
    const float* mean0 = (const float*)d_in[2];
    const float* cov0  = (const float*)d_in[3];
    float* out = (float*)d_out;
    float* ws  = (float*)d_ws;

    const int NE = DD * BB;                 // 65536 elements per matrix
    float* r   = ws;
    float* xs  = ws + 1 * NE;
    float* dA  = ws + 2 * NE;
    float* dB  = ws + 3 * NE;
    float* a2  = ws + 4 * NE;
    float* s2  = a2 + BB;
    float* c1t = s2 + BB;
    float* c2t = c1t + NITER * BB;

    coef_kernel<<<1, BB, 0, stream>>>(t, a2, s2, c1t, c2t);
    init_kernel<<<NE / 256, 256, 0, stream>>>(x, mean0, a2, s2, r, dA, xs);

    float* dbuf[2] = {dA, dB};
    for (int k = 1; k < NITER; ++k) {
        cheb_iter_kernel<<<dim3(DD / 16, 2), 128, 0, stream>>>(
            cov0, dbuf[(k - 1) & 1], dbuf[k & 1], r, xs, a2, s2,
            c1t + k * BB, c2t + k * BB);
    }
    finalize_kernel<<<NE / 256, 256, 0, stream>>>(xs, out);
}